// ssim_criterion_2430951489994
// MI455X (gfx1250) — compile-verified
//
#include <hip/hip_runtime.h>

// ---------------------------------------------------------------------------
// Fused SSIM loss for (32,3,512,512) fp32, MI455X / gfx1250.
//
// Memory-bound: 201 MB input, scalar output -> single fused pass (~9us HBM
// floor at 23.3 TB/s). Horizontal Gaussian blur runs on the XDL pipe as
// chained V_WMMA_F32_16X16X4_F32 (D = V[16x28] x bandmatrix W[28x16]);
// vertical blur + SSIM epilogue on VALU; one float atomic per block.
// B-fragments (band weights) are synthesized branchlessly with v_exp_f32 and
// hoisted out of the WMMA chunk loop so the chain is ds_load + wmma only.
// ---------------------------------------------------------------------------

typedef float v2f __attribute__((ext_vector_type(2)));
typedef float v8f __attribute__((ext_vector_type(8)));

#define BATCH    32
#define CHANS    3
#define NPLANES  (BATCH * CHANS)   // 96
#define H_IN     512
#define W_IN     512
#define WIN      11
#define OUT_H    502               // 512 - 10 (VALID)
#define OUT_W    502
#define ROW_TILE 16
#define COL_TILE 128               // 8 waves x 16 cols
#define IN_ROWS  26                // ROW_TILE + WIN - 1
#define IN_COLS  140               // COL_TILE + 12 (K padded 26 -> 28)
#define LDS_PITCH 141              // odd pitch: bank-conflict padding
#define N_ROW_TILES 32             // ceil(502/16)
#define N_COL_TILES 4              // ceil(502/128)

// Normalized 11-tap Gaussian, sigma = 1.5 (pytorch-msssim window).
// constexpr -> folded to inline literals (no .rodata scalar loads).
constexpr float GW[WIN] = {
    0.00102840f, 0.00759876f, 0.03600075f, 0.10936066f, 0.21300557f,
    0.26601160f,
    0.21300557f, 0.10936066f, 0.03600075f, 0.00759876f, 0.00102840f};

// Branchless band-matrix tap: w(d) = exp(-(d-5)^2 / (2*1.5^2)) / sum,
// zero outside [0,10]. exp via v_exp_f32: exp(-x) = 2^(-x*log2e).
__device__ __forceinline__ float gw_tap(int d) {
  const float t = (float)(d - 5);
  // log2(e)/4.5 = 0.320598898 ; 1/sum = 0.26601160
  const float val =
      __builtin_amdgcn_exp2f(t * t * -0.320598898f) * 0.26601160f;
  return ((unsigned)d < (unsigned)WIN) ? val : 0.0f;
}

__global__ __launch_bounds__(256) void ssim_fused_kernel(
    const float* __restrict__ X, const float* __restrict__ Y,
    float* __restrict__ plane_acc) {
  // --- block -> (plane, row tile, col group) -------------------------------
  const int blk   = blockIdx.x;
  const int colg  = blk % N_COL_TILES;
  const int rowg  = (blk / N_COL_TILES) % N_ROW_TILES;
  const int plane = blk / (N_COL_TILES * N_ROW_TILES);
  const int r0 = rowg * ROW_TILE;
  const int c0 = colg * COL_TILE;
  const float* __restrict__ xp = X + (size_t)plane * (H_IN * W_IN);
  const float* __restrict__ yp = Y + (size_t)plane * (H_IN * W_IN);

  __shared__ float sx[IN_ROWS][LDS_PITCH];            // normalized x halo tile
  __shared__ float sy[IN_ROWS][LDS_PITCH];            // normalized y halo tile
  __shared__ float sv[5][ROW_TILE][LDS_PITCH];        // vertically blurred q's
  __shared__ float sred[8];

  const int tid = threadIdx.x;

  // --- Phase A: global -> LDS (coalesced, clamped at image edge) -----------
  for (int i = tid; i < IN_ROWS * IN_COLS; i += 256) {
    const int r = i / IN_COLS, c = i % IN_COLS;
    const int gr = min(r0 + r, H_IN - 1);
    const int gc = min(c0 + c, W_IN - 1);
    const float xv = (xp[gr * W_IN + gc] + 1.0f) * 0.5f;   // (-1,1) -> (0,1)
    const float yv = (yp[gr * W_IN + gc] + 1.0f) * 0.5f;
    sx[r][c] = xv;
    sy[r][c] = yv;
  }
  __syncthreads();

  // --- Phase B: vertical blur of {x, y, x^2, y^2, xy} (VALU) ---------------
  for (int i = tid; i < ROW_TILE * IN_COLS; i += 256) {
    const int r = i / IN_COLS, c = i % IN_COLS;
    float ax = 0.f, ay = 0.f, axx = 0.f, ayy = 0.f, axy = 0.f;
#pragma unroll
    for (int k = 0; k < WIN; ++k) {
      const float wk = GW[k];
      const float xv = sx[r + k][c];
      const float yv = sy[r + k][c];
      ax  += wk * xv;       ay  += wk * yv;
      axx += wk * xv * xv;  ayy += wk * yv * yv;  axy += wk * xv * yv;
    }
    sv[0][r][c] = ax;  sv[1][r][c] = ay;
    sv[2][r][c] = axx; sv[3][r][c] = ayy; sv[4][r][c] = axy;
  }
  __syncthreads();

  // --- Phase C: horizontal blur as WMMA, one 16x16 tile per wave -----------
  // D(16x16) = A(16x28) x W(28x16), W[k][c] = GW[k-c] (banded), K chunked by 4.
  const int wave = tid >> 5;
  const int lane = tid & 31;
  const int half = lane >> 4;     // A: K-pair select, B: K-row select
  const int l15  = lane & 15;     // A: row M,        B: col N
  const int colbase = wave * 16;

  // Hoist all 7 B fragments (band weights) out of the WMMA chain: branchless
  // v_exp_f32 + cndmask, no divergent loads between WMMAs.
  v2f bf[7];
#pragma unroll
  for (int kc = 0; kc < 7; ++kc) {
    const int kB = kc * 4 + half * 2;      // this lane's two K indices
    bf[kc].x = gw_tap(kB - l15);
    bf[kc].y = gw_tap(kB + 1 - l15);
  }

  v8f dacc[5];
#pragma unroll
  for (int q = 0; q < 5; ++q) {
    v8f z = {0.f, 0.f, 0.f, 0.f, 0.f, 0.f, 0.f, 0.f};
    dacc[q] = z;
  }

#pragma unroll
  for (int kc = 0; kc < 7; ++kc) {
    const int kB = kc * 4 + half * 2;
#pragma unroll
    for (int q = 0; q < 5; ++q) {
      v2f afrag;          // A 16x4 f32: lanes 0-15 hold K=kB..kB+1 of row l15
      afrag.x = sv[q][l15][colbase + kB];
      afrag.y = sv[q][l15][colbase + kB + 1];
      dacc[q] = __builtin_amdgcn_wmma_f32_16x16x4_f32(
          false, afrag, false, bf[kc], (short)0, dacc[q], false, false);
    }
  }

  // --- Phase D: SSIM map on D layout, masked partial sum -------------------
  const float c1v = 0.0001f;   // (K1*range)^2
  const float c2v = 0.0009f;   // (K2*range)^2
  float lsum = 0.0f;
  const int gcol0 = c0 + colbase;
#pragma unroll
  for (int i = 0; i < 8; ++i) {
    const int m  = i + half * 8;     // D VGPR i: M=i (lanes 0-15), M=i+8 (16-31)
    const int gr = r0 + m;
    const int gc = gcol0 + l15;
    if (gr < OUT_H && gc < OUT_W) {
      const float mu1 = dacc[0][i], mu2 = dacc[1][i];
      const float s11 = dacc[2][i] - mu1 * mu1;
      const float s22 = dacc[3][i] - mu2 * mu2;
      const float s12 = dacc[4][i] - mu1 * mu2;
      const float cs   = (2.0f * s12 + c2v) / (s11 + s22 + c2v);
      const float ssim = ((2.0f * mu1 * mu2 + c1v) /
                          (mu1 * mu1 + mu2 * mu2 + c1v)) * cs;
      lsum += ssim;
    }
  }
#pragma unroll
  for (int ofs = 16; ofs > 0; ofs >>= 1)
    lsum += __shfl_down(lsum, ofs, 32);
  if (lane == 0) sred[wave] = lsum;
  __syncthreads();
  if (tid == 0) {
    float s = 0.f;
#pragma unroll
    for (int wv = 0; wv < 8; ++wv) s += sred[wv];
    atomicAdd(&plane_acc[plane], s);
  }
}

__global__ void ssim_init_kernel(float* __restrict__ plane_acc) {
  if (threadIdx.x < NPLANES) plane_acc[threadIdx.x] = 0.0f;
}

__global__ void ssim_final_kernel(const float* __restrict__ plane_acc,
                                  float* __restrict__ out) {
  __shared__ float s[128];
  const int t = threadIdx.x;
  float v = 0.0f;
  if (t < NPLANES) {
    const float m = plane_acc[t] * (1.0f / (float)(OUT_H * OUT_W));
    v = m > 0.0f ? m : 0.0f;                 // nonnegative_ssim (relu)
  }
  s[t] = v;
  __syncthreads();
  for (int ofs = 64; ofs > 0; ofs >>= 1) {
    if (t < ofs) s[t] += s[t + ofs];
    __syncthreads();
  }
  if (t == 0) out[0] = 1.0f - s[0] * (1.0f / (float)NPLANES);
}

extern "C" void kernel_launch(void* const* d_in, const int* in_sizes, int n_in,
                              void* d_out, int out_size, void* d_ws,
                              size_t ws_size, hipStream_t stream) {
  const float* X = (const float*)d_in[0];
  const float* Y = (const float*)d_in[1];
  float* acc = (float*)d_ws;           // 96 fp32 per-plane accumulators
  float* out = (float*)d_out;

  ssim_init_kernel<<<1, 128, 0, stream>>>(acc);
  const int nblocks = NPLANES * N_ROW_TILES * N_COL_TILES;   // 12288
  ssim_fused_kernel<<<nblocks, 256, 0, stream>>>(X, Y, acc);
  ssim_final_kernel<<<1, 128, 0, stream>>>(acc, out);
}